// GAT_18554258719054
// MI455X (gfx1250) — compile-verified
//
#include <hip/hip_runtime.h>
#include <hip/hip_bf16.h>

typedef __attribute__((ext_vector_type(2))) float v2f;
typedef __attribute__((ext_vector_type(8))) float v8f;

#define DIM_IN 128
#define HEADS 15
#define DIM_H 8
#define HC 120           // HEADS*DIM_H
#define DIM_OUT 32
#define NEG_SLOPE 0.2f

// ---------- helpers ----------
__device__ __forceinline__ unsigned ordEnc(float f) {
    unsigned u = __float_as_uint(f);
    return (u & 0x80000000u) ? ~u : (u | 0x80000000u);
}
__device__ __forceinline__ float ordDec(unsigned u) {
    return __uint_as_float((u & 0x80000000u) ? (u ^ 0x80000000u) : ~u);
}
__device__ __forceinline__ float lrelu(float v) {
    return v > 0.0f ? v : NEG_SLOPE * v;
}
__device__ __forceinline__ void edge_endpoints(int e, int E, const long long* src,
                                               const long long* dst, int& s, int& d) {
    if (e < E) { s = (int)src[e]; d = (int)dst[e]; }
    else       { s = d = e - E; }            // self-loops appended after real edges
}

// layer-1 attention logit for (src s, dst d, head h); also returns xl[src] row
__device__ __forceinline__ float logit1(const float* __restrict__ xl,
                                        const float* __restrict__ xr,
                                        const float* __restrict__ att,
                                        int s, int d, int h,
                                        float4& l0, float4& l1) {
    const float4* pl = (const float4*)(xl + (size_t)s * HC + h * DIM_H);
    const float4* pr = (const float4*)(xr + (size_t)d * HC + h * DIM_H);
    const float4* pa = (const float4*)(att + h * DIM_H);
    l0 = pl[0]; l1 = pl[1];
    float4 r0 = pr[0], r1 = pr[1], a0 = pa[0], a1 = pa[1];
    return lrelu(l0.x + r0.x) * a0.x + lrelu(l0.y + r0.y) * a0.y
         + lrelu(l0.z + r0.z) * a0.z + lrelu(l0.w + r0.w) * a0.w
         + lrelu(l1.x + r1.x) * a1.x + lrelu(l1.y + r1.y) * a1.y
         + lrelu(l1.z + r1.z) * a1.z + lrelu(l1.w + r1.w) * a1.w;
}

// ---------- fp32 WMMA GEMM: C[M,Ncol] = A[M,K] @ B[K,Ncol], row-major ----------
// One wave per 16x16 tile, K in steps of 4 via v_wmma_f32_16x16x4_f32.
// M multiple of 16 (50000 = 3125*16). Out-of-range columns handled branch-free
// via clamped address + zero mask, so the K-loop has no EXEC manipulation.
__global__ __launch_bounds__(256) void wmma_gemm_f32(
    const float* __restrict__ A, const float* __restrict__ B, float* __restrict__ C,
    int M, int K, int Ncol)
{
    const int wave = threadIdx.x >> 5;
    const int lane = threadIdx.x & 31;
    const int tileM = blockIdx.x * 8 + wave;
    if (tileM * 16 >= M) return;                    // uniform per wave

    const int row0 = tileM * 16;
    const int col0 = blockIdx.y * 16;
    const int aRow = lane & 15;                     // A: both half-waves hold M=0..15
    const int kPair = (lane >> 4) << 1;             // K sub-offset: 0 or 2
    const int c = col0 + (lane & 15);
    const bool cOk = (c < Ncol);
    const int cc = cOk ? c : (Ncol - 1);            // clamped (always valid) column
    const float msk = cOk ? 1.0f : 0.0f;

    const float* __restrict__ Arow = A + (size_t)(row0 + aRow) * K + kPair;
    const float* __restrict__ Bcol = B + (size_t)kPair * Ncol + cc;

    v8f acc = {};
    #pragma unroll 4
    for (int k = 0; k < K; k += 4) {
        v2f a, b;
        a.x = Arow[k];
        a.y = Arow[k + 1];
        b.x = msk * Bcol[(size_t)k * Ncol];
        b.y = msk * Bcol[(size_t)(k + 1) * Ncol];
        acc = __builtin_amdgcn_wmma_f32_16x16x4_f32(
            false, a, false, b, (short)0, acc, false, false);
    }

    // C/D layout: VGPR g -> row g (lanes 0-15) / row g+8 (lanes 16-31)
    const int rBase = row0 + ((lane >> 4) << 3);
    if (cOk) {
        #pragma unroll
        for (int g = 0; g < 8; ++g)
            C[(size_t)(rBase + g) * Ncol + c] = acc[g];
    }
}

// ---------- layer 1: pass A = logits + segment max (thread = edge*HEADS+head) ----------
__global__ __launch_bounds__(256) void edge_max1(
    const float* __restrict__ xl, const float* __restrict__ xr,
    const long long* __restrict__ src, const long long* __restrict__ dst,
    const float* __restrict__ att, unsigned* __restrict__ emax, int E, int Et)
{
    int t = blockIdx.x * blockDim.x + threadIdx.x;
    if (t >= Et * HEADS) return;
    int e = t / HEADS, h = t - e * HEADS;
    int s, d; edge_endpoints(e, E, src, dst, s, d);
    float4 l0, l1;
    float v = logit1(xl, xr, att, s, d, h, l0, l1);
    atomicMax(&emax[(size_t)d * HEADS + h], ordEnc(v));
}

// ---------- layer 1: pass B = exp + denom + numerator accumulation ----------
__global__ __launch_bounds__(256) void edge_acc1(
    const float* __restrict__ xl, const float* __restrict__ xr,
    const long long* __restrict__ src, const long long* __restrict__ dst,
    const float* __restrict__ att, const unsigned* __restrict__ emax,
    float* __restrict__ denom, float* __restrict__ num, int E, int Et)
{
    int t = blockIdx.x * blockDim.x + threadIdx.x;
    if (t >= Et * HEADS) return;
    int e = t / HEADS, h = t - e * HEADS;
    int s, d; edge_endpoints(e, E, src, dst, s, d);
    float4 l0, l1;
    float v = logit1(xl, xr, att, s, d, h, l0, l1);
    float ee = __expf(v - ordDec(emax[(size_t)d * HEADS + h]));

    atomicAdd(&denom[(size_t)d * HEADS + h], ee);
    float* od = num + (size_t)d * HC + h * DIM_H;
    atomicAdd(&od[0], ee * l0.x);  atomicAdd(&od[1], ee * l0.y);
    atomicAdd(&od[2], ee * l0.z);  atomicAdd(&od[3], ee * l0.w);
    atomicAdd(&od[4], ee * l1.x);  atomicAdd(&od[5], ee * l1.y);
    atomicAdd(&od[6], ee * l1.z);  atomicAdd(&od[7], ee * l1.w);
}

// ---------- layer 1: normalize + bias + ELU ----------
__global__ __launch_bounds__(256) void norm_bias_elu1(
    const float* __restrict__ num, const float* __restrict__ denom,
    const float* __restrict__ b1, float* __restrict__ h1, int total)
{
    int t = blockIdx.x * blockDim.x + threadIdx.x;
    if (t >= total) return;
    int node = t / HC, col = t - node * HC;
    float v = num[t] / denom[(size_t)node * HEADS + (col >> 3)] + b1[col];
    h1[t] = v > 0.0f ? v : (__expf(v) - 1.0f);
}

// ---------- layer 2 (H=1, C=32) ----------
__device__ __forceinline__ float logit2(const float* __restrict__ hl,
                                        const float* __restrict__ hr,
                                        const float* __restrict__ att2,
                                        int s, int d, float4* lsave) {
    const float4* pl = (const float4*)(hl + (size_t)s * DIM_OUT);
    const float4* pr = (const float4*)(hr + (size_t)d * DIM_OUT);
    const float4* pa = (const float4*)att2;
    float v = 0.0f;
    #pragma unroll
    for (int g = 0; g < DIM_OUT / 4; ++g) {
        float4 l = pl[g], r = pr[g], a = pa[g];
        lsave[g] = l;
        v += lrelu(l.x + r.x) * a.x + lrelu(l.y + r.y) * a.y
           + lrelu(l.z + r.z) * a.z + lrelu(l.w + r.w) * a.w;
    }
    return v;
}

__global__ __launch_bounds__(256) void edge_max2(
    const float* __restrict__ hl, const float* __restrict__ hr,
    const long long* __restrict__ src, const long long* __restrict__ dst,
    const float* __restrict__ att2, unsigned* __restrict__ emax, int E, int Et)
{
    int e = blockIdx.x * blockDim.x + threadIdx.x;
    if (e >= Et) return;
    int s, d; edge_endpoints(e, E, src, dst, s, d);
    float4 lv[DIM_OUT / 4];
    float v = logit2(hl, hr, att2, s, d, lv);
    atomicMax(&emax[d], ordEnc(v));
}

__global__ __launch_bounds__(256) void edge_acc2(
    const float* __restrict__ hl, const float* __restrict__ hr,
    const long long* __restrict__ src, const long long* __restrict__ dst,
    const float* __restrict__ att2, const unsigned* __restrict__ emax,
    float* __restrict__ denom, float* __restrict__ num, int E, int Et)
{
    int e = blockIdx.x * blockDim.x + threadIdx.x;
    if (e >= Et) return;
    int s, d; edge_endpoints(e, E, src, dst, s, d);
    float4 lv[DIM_OUT / 4];
    float v = logit2(hl, hr, att2, s, d, lv);
    float ee = __expf(v - ordDec(emax[d]));

    atomicAdd(&denom[d], ee);
    float* od = num + (size_t)d * DIM_OUT;
    #pragma unroll
    for (int g = 0; g < DIM_OUT / 4; ++g) {
        atomicAdd(&od[g * 4 + 0], ee * lv[g].x);
        atomicAdd(&od[g * 4 + 1], ee * lv[g].y);
        atomicAdd(&od[g * 4 + 2], ee * lv[g].z);
        atomicAdd(&od[g * 4 + 3], ee * lv[g].w);
    }
}

// ---------- finalize: normalize + bias, write h2, wave32 log_softmax ----------
__global__ __launch_bounds__(256) void finalize(
    const float* __restrict__ num, const float* __restrict__ denom,
    const float* __restrict__ b2, float* __restrict__ out, int Nn)
{
    int node = blockIdx.x * 8 + (threadIdx.x >> 5);
    int lane = threadIdx.x & 31;
    if (node >= Nn) return;

    float v = num[(size_t)node * DIM_OUT + lane] / denom[node] + b2[lane];
    out[(size_t)node * DIM_OUT + lane] = v;

    float m = v;
    #pragma unroll
    for (int off = 16; off > 0; off >>= 1)
        m = fmaxf(m, __shfl_xor(m, off, 32));
    float sum = __expf(v - m);
    #pragma unroll
    for (int off = 16; off > 0; off >>= 1)
        sum += __shfl_xor(sum, off, 32);
    out[(size_t)Nn * DIM_OUT + (size_t)node * DIM_OUT + lane] = (v - m) - __logf(sum);
}

// ---------- host side ----------
extern "C" void kernel_launch(void* const* d_in, const int* in_sizes, int n_in,
                              void* d_out, int out_size, void* d_ws, size_t ws_size,
                              hipStream_t stream) {
    const float*     x    = (const float*)d_in[0];
    const long long* ei   = (const long long*)d_in[1];
    const float*     W1l  = (const float*)d_in[2];
    const float*     W1r  = (const float*)d_in[3];
    const float*     att1 = (const float*)d_in[4];
    const float*     b1   = (const float*)d_in[5];
    const float*     W2l  = (const float*)d_in[6];
    const float*     W2r  = (const float*)d_in[7];
    const float*     att2 = (const float*)d_in[8];
    const float*     b2   = (const float*)d_in[9];
    float* out = (float*)d_out;

    const int Nn = in_sizes[0] / DIM_IN;   // 50000
    const int E  = in_sizes[1] / 2;        // 800000
    const int Et = E + Nn;                 // with self-loops
    const long long* srcI = ei;
    const long long* dstI = ei + E;

    // ---- workspace layout (floats); ~122 MB total, no aliasing needed ----
    float* ws = (float*)d_ws;
    float*    xl     = ws;                                   // N*120
    float*    xr     = xl + (size_t)Nn * HC;                 // N*120
    float*    num1   = xr + (size_t)Nn * HC;                 // N*120 (numerator accum)
    float*    h1     = num1 + (size_t)Nn * HC;               // N*120 (layer-1 output)
    unsigned* emax1  = (unsigned*)(h1 + (size_t)Nn * HC);    // N*15
    float*    denom1 = (float*)(emax1 + (size_t)Nn * HEADS); // N*15
    float*    hl2    = denom1 + (size_t)Nn * HEADS;          // N*32
    float*    hr2    = hl2 + (size_t)Nn * DIM_OUT;           // N*32
    float*    num2   = hr2 + (size_t)Nn * DIM_OUT;           // N*32
    unsigned* emax2  = (unsigned*)(num2 + (size_t)Nn * DIM_OUT); // N
    float*    denom2 = (float*)(emax2 + Nn);                 // N

    const int TPB = 256;
    const int mTiles = Nn / 16;                              // 3125
    dim3 gemmBlk(256);

    // ---- layer 1 ----
    hipMemsetAsync(num1, 0, (size_t)Nn * HC * sizeof(float), stream);
    hipMemsetAsync(emax1, 0, (size_t)Nn * HEADS * sizeof(unsigned), stream);
    hipMemsetAsync(denom1, 0, (size_t)Nn * HEADS * sizeof(float), stream);

    dim3 g1((mTiles + 7) / 8, (HC + 15) / 16);
    wmma_gemm_f32<<<g1, gemmBlk, 0, stream>>>(x, W1l, xl, Nn, DIM_IN, HC);
    wmma_gemm_f32<<<g1, gemmBlk, 0, stream>>>(x, W1r, xr, Nn, DIM_IN, HC);

    int eh = Et * HEADS;
    edge_max1<<<(eh + TPB - 1) / TPB, TPB, 0, stream>>>(xl, xr, srcI, dstI, att1,
                                                        emax1, E, Et);
    edge_acc1<<<(eh + TPB - 1) / TPB, TPB, 0, stream>>>(xl, xr, srcI, dstI, att1,
                                                        emax1, denom1, num1, E, Et);
    int nh = Nn * HC;
    norm_bias_elu1<<<(nh + TPB - 1) / TPB, TPB, 0, stream>>>(num1, denom1, b1, h1, nh);

    // ---- layer 2 ----
    hipMemsetAsync(num2, 0, (size_t)Nn * DIM_OUT * sizeof(float), stream);
    hipMemsetAsync(emax2, 0, (size_t)Nn * sizeof(unsigned), stream);
    hipMemsetAsync(denom2, 0, (size_t)Nn * sizeof(float), stream);

    dim3 g2((mTiles + 7) / 8, (DIM_OUT + 15) / 16);
    wmma_gemm_f32<<<g2, gemmBlk, 0, stream>>>(h1, W2l, hl2, Nn, HC, DIM_OUT);
    wmma_gemm_f32<<<g2, gemmBlk, 0, stream>>>(h1, W2r, hr2, Nn, HC, DIM_OUT);

    edge_max2<<<(Et + TPB - 1) / TPB, TPB, 0, stream>>>(hl2, hr2, srcI, dstI, att2,
                                                        emax2, E, Et);
    edge_acc2<<<(Et + TPB - 1) / TPB, TPB, 0, stream>>>(hl2, hr2, srcI, dstI, att2,
                                                        emax2, denom2, num2, E, Et);

    finalize<<<(Nn + 7) / 8, TPB, 0, stream>>>(num2, denom2, b2, out, Nn);
}